// ChamferDistance_83451214561689
// MI455X (gfx1250) — compile-verified
//
#include <hip/hip_runtime.h>
#include <hip/hip_bf16.h>

typedef __attribute__((ext_vector_type(2))) float v2f;
typedef __attribute__((ext_vector_type(8))) float v8f;

#define CHUNK 1024          // reference points staged in LDS per pass (12 KB)
#define THREADS 256         // 8 waves (wave32)
#define ROWS_PER_BLOCK 128  // 8 waves * 16 query rows
#define IDX_MASK 0x1FFFu    // 13 bits: M up to 8192

// For each query point, min squared distance + argmin over all ref points.
// WMMA computes  d2 = A x B + C  with A row = [-2ax,-2ay,-2az,1],
// B col = [bx,by,bz,|b|^2], C[row,*] = |a_row|^2  ->  full d2 >= 0 per tile.
// (dist,idx) packed into one float key: low 13 mantissa bits replaced by the
// candidate index; since d2 >= 0, float ordering == packed ordering, and ties
// in the quantized distance pick the smaller index (argmin first-occurrence).
__global__ __launch_bounds__(THREADS)
void chamfer_nn_wmma(const float* __restrict__ q,   // (B,Nq,3)
                     const float* __restrict__ r,   // (B,Mr,3)
                     float* __restrict__ dist_out,  // (B,Nq)
                     float* __restrict__ idx_out,   // (B,Nq) as float
                     int Nq, int Mr)
{
    __shared__ float s[3 * CHUNK];   // SoA: x[CHUNK], y[CHUNK], z[CHUNK]

    const int b    = blockIdx.y;
    const int tid  = threadIdx.x;
    const int wave = tid >> 5;
    const int lane = tid & 31;
    const int col  = lane & 15;
    const bool hi  = lane >= 16;

    const int n0 = blockIdx.x * ROWS_PER_BLOCK + wave * 16;
    if (n0 >= Nq) return;

    // This wave's 16 query points (row = n0+col; both halves load same rows).
    int qrow = n0 + col;
    if (qrow >= Nq) qrow = Nq - 1;
    const float* qp = q + ((long)b * Nq + qrow) * 3;
    const float ax = qp[0], ay = qp[1], az = qp[2];
    const float an2 = ax*ax + ay*ay + az*az;

    // A-matrix 16x4 f32: lanes 0-15 -> K=0 (v.x), K=1 (v.y);
    //                    lanes 16-31 -> K=2 (v.x), K=3 (v.y).
    v2f areg;
    areg.x = hi ? (-2.0f * az) : (-2.0f * ax);
    areg.y = hi ? 1.0f         : (-2.0f * ay);

    // C input: per-row query norms. C layout: VGPR v, lanes 0-15 row v,
    // lanes 16-31 row v+8; row k's norm lives in lane k.
    v8f cin;
    #pragma unroll
    for (int v = 0; v < 8; ++v)
        cin[v] = __shfl(an2, v + (hi ? 8 : 0), 32);

    float best[8];
    #pragma unroll
    for (int v = 0; v < 8; ++v) best[v] = INFINITY;

    const float* rb = r + (long)b * Mr * 3;

    for (int mb = 0; mb < Mr; mb += CHUNK) {
        __syncthreads();
        // Cooperative coalesced AoS load, SoA store into LDS.
        for (int off = tid; off < 3 * CHUNK; off += THREADS) {
            float vv = rb[(long)mb * 3 + off];
            int pt = off / 3;
            int d  = off - pt * 3;
            s[d * CHUNK + pt] = vv;
        }
        // Prefetch next chunk into cache while this one is consumed.
        if (mb + CHUNK < Mr)
            __builtin_prefetch(rb + (long)(mb + CHUNK) * 3 + tid, 0, 1);
        __syncthreads();

        #pragma unroll 4
        for (int j = 0; j < CHUNK; j += 32) {       // two 16-col tiles / iter
            const int i0 = j + col;
            const int i1 = i0 + 16;
            const float bx0 = s[i0],             bx1 = s[i1];
            const float by0 = s[CHUNK + i0],     by1 = s[CHUNK + i1];
            const float bz0 = s[2 * CHUNK + i0], bz1 = s[2 * CHUNK + i1];
            const float bn0 = bx0*bx0 + by0*by0 + bz0*bz0;
            const float bn1 = bx1*bx1 + by1*by1 + bz1*bz1;

            v2f breg0, breg1;
            breg0.x = hi ? bz0 : bx0;  breg0.y = hi ? bn0 : by0;
            breg1.x = hi ? bz1 : bx1;  breg1.y = hi ? bn1 : by1;

            v8f c0 = __builtin_amdgcn_wmma_f32_16x16x4_f32(
                         false, areg, false, breg0, (short)0, cin, false, false);
            v8f c1 = __builtin_amdgcn_wmma_f32_16x16x4_f32(
                         false, areg, false, breg1, (short)0, cin, false, false);

            const unsigned cand0 = (unsigned)(mb + j + col);
            const unsigned cand1 = cand0 + 16u;
            #pragma unroll
            for (int v = 0; v < 8; ++v) {
                float k0 = __uint_as_float(
                    (__float_as_uint(c0[v]) & ~IDX_MASK) | cand0);  // v_and_or
                float k1 = __uint_as_float(
                    (__float_as_uint(c1[v]) & ~IDX_MASK) | cand1);
                best[v] = fminf(fminf(k0, k1), best[v]);            // v_min3
            }
        }
    }

    // Reduce packed keys over the 16 columns; xor masks {8,4,2,1} stay
    // within each wave32 half (low half: rows v, high half: rows v+8).
    #pragma unroll
    for (int v = 0; v < 8; ++v) {
        #pragma unroll
        for (int off = 8; off >= 1; off >>= 1)
            best[v] = fminf(best[v], __shfl_xor(best[v], off, 32));
    }

    if (col == 0) {  // lanes 0 (rows n0+0..7) and 16 (rows n0+8..15) write
        const int rowbase = n0 + (hi ? 8 : 0);
        const long obase  = (long)b * Nq + rowbase;
        #pragma unroll
        for (int v = 0; v < 8; ++v) {
            const int row = rowbase + v;
            if (row < Nq) {
                const int idx = (int)(__float_as_uint(best[v]) & IDX_MASK);
                // Exact fp32 recompute of the winning pair's distance.
                const float* ap = q + ((long)b * Nq + row) * 3;
                const float* bp = rb + (long)idx * 3;
                const float dx = ap[0] - bp[0];
                const float dy = ap[1] - bp[1];
                const float dz = ap[2] - bp[2];
                dist_out[obase + v] = dx*dx + dy*dy + dz*dz;
                idx_out[obase + v]  = (float)idx;
            }
        }
    }
}

extern "C" void kernel_launch(void* const* d_in, const int* in_sizes, int n_in,
                              void* d_out, int out_size, void* d_ws, size_t ws_size,
                              hipStream_t stream) {
    const float* xyz1 = (const float*)d_in[0];
    const float* xyz2 = (const float*)d_in[1];

    const int D = 3, B = 4;
    const int N = in_sizes[0] / (B * D);
    const int M = in_sizes[1] / (B * D);

    float* out   = (float*)d_out;
    float* dist1 = out;                                   // (B,N)
    float* dist2 = out + (long)B * N;                     // (B,M)
    float* idx1  = out + (long)B * N + (long)B * M;       // (B,N)
    float* idx2  = out + 2L * B * N + (long)B * M;        // (B,M)

    dim3 block(THREADS);

    // Pass 1: for each xyz1 point, nearest in xyz2.
    dim3 grid1((N + ROWS_PER_BLOCK - 1) / ROWS_PER_BLOCK, B);
    chamfer_nn_wmma<<<grid1, block, 0, stream>>>(xyz1, xyz2, dist1, idx1, N, M);

    // Pass 2: roles swapped — for each xyz2 point, nearest in xyz1.
    dim3 grid2((M + ROWS_PER_BLOCK - 1) / ROWS_PER_BLOCK, B);
    chamfer_nn_wmma<<<grid2, block, 0, stream>>>(xyz2, xyz1, dist2, idx2, M, N);
}